// NoBrainEncoderBlock_25555055411290
// MI455X (gfx1250) — compile-verified
//
#include <hip/hip_runtime.h>
#include <math.h>

typedef __attribute__((ext_vector_type(2))) float v2f;
typedef __attribute__((ext_vector_type(8))) float v8f;

#define DQ 128
#define TOPK 25
#define CHUNK 4096

// ---------------------------------------------------------------------------
// Kernel 0: header init — inverse q norms and sigmoid gate
// hdr[0]=Z1 hdr[1]=Z2 hdr[2]=1/|q1| hdr[3]=1/|q2| hdr[4]=a
// ---------------------------------------------------------------------------
__global__ void nb_init_kernel(const float* __restrict__ q1,
                               const float* __restrict__ q2,
                               const float* __restrict__ temp,
                               float* __restrict__ hdr) {
  __shared__ float r[DQ];
  int t = threadIdx.x;  // blockDim = 128
  float v = q1[t];
  r[t] = v * v;
  __syncthreads();
  for (int off = 64; off > 0; off >>= 1) {
    if (t < off) r[t] += r[t + off];
    __syncthreads();
  }
  float ss1 = r[0];
  __syncthreads();
  v = q2[t];
  r[t] = v * v;
  __syncthreads();
  for (int off = 64; off > 0; off >>= 1) {
    if (t < off) r[t] += r[t + off];
    __syncthreads();
  }
  if (t == 0) {
    hdr[0] = 0.0f;
    hdr[1] = 0.0f;
    hdr[2] = 1.0f / fmaxf(sqrtf(ss1), 1e-12f);
    hdr[3] = 1.0f / fmaxf(sqrtf(r[0]), 1e-12f);
    hdr[4] = 1.0f / (1.0f + expf(-temp[0]));
  }
}

// ---------------------------------------------------------------------------
// Kernel 1: WMMA fp32 scores. One wave = one tile of 16 keys.
// A (16x4) = q chunk replicated over all 16 rows -> every row of D equals
// dot(q, key_n).  B (4x16) = 16 keys' 4-dim chunk.  Accumulate K=128 in 32
// steps of V_WMMA_F32_16X16X4_F32.  Key sum-of-squares via VALU alongside.
// ---------------------------------------------------------------------------
__global__ void nb_scores_kernel(const float* __restrict__ q1,
                                 const float* __restrict__ k1,
                                 const float* __restrict__ q2,
                                 const float* __restrict__ k2,
                                 const float* __restrict__ hdr,
                                 float* __restrict__ e1,
                                 float* __restrict__ e2,
                                 float* __restrict__ partials,
                                 int N) {
  __shared__ float lq1[DQ];
  __shared__ float lq2[DQ];
  __shared__ float wred1[8];
  __shared__ float wred2[8];

  int t = threadIdx.x;              // blockDim = 256 (8 waves)
  if (t < DQ) lq1[t] = q1[t];
  else        lq2[t - DQ] = q2[t - DQ];
  __syncthreads();

  float invq1 = hdr[2];
  float invq2 = hdr[3];

  int lane  = t & 31;
  int wave  = t >> 5;
  int laneN = lane & 15;            // key within tile (B column / D column)
  int hi    = lane >> 4;            // lane half selects K sub-pair
  int kAdj  = hi * 2;

  int numTiles = (N + 15) / 16;
  int tile = blockIdx.x * 8 + wave; // wave-uniform

  float z1 = 0.0f, z2 = 0.0f;

  if (tile < numTiles) {            // uniform branch: EXEC all-ones inside
    int key0 = tile * 16 + laneN;
    int key  = key0 < N ? key0 : (N - 1);
    const float* __restrict__ r1 = k1 + (size_t)key * DQ;
    const float* __restrict__ r2 = k2 + (size_t)key * DQ;

    v8f c1 = {};
    v8f c2 = {};
    float ss1 = 0.0f, ss2 = 0.0f;

#pragma unroll
    for (int s = 0; s < DQ / 4; ++s) {
      int kb = s * 4 + kAdj;
      v2f a1 = { lq1[kb], lq1[kb + 1] };   // A row (same for all M)
      v2f a2 = { lq2[kb], lq2[kb + 1] };
      v2f b1 = *(const v2f*)(r1 + kb);     // contiguous 8B key chunk
      v2f b2 = *(const v2f*)(r2 + kb);
      ss1 = fmaf(b1.x, b1.x, fmaf(b1.y, b1.y, ss1));
      ss2 = fmaf(b2.x, b2.x, fmaf(b2.y, b2.y, ss2));
      c1 = __builtin_amdgcn_wmma_f32_16x16x4_f32(false, a1, false, b1,
                                                 (short)0, c1, false, false);
      c2 = __builtin_amdgcn_wmma_f32_16x16x4_f32(false, a2, false, b2,
                                                 (short)0, c2, false, false);
    }

    // lane l saw dims K = {4t+kAdj, 4t+kAdj+1}; partner lane l^16 has the rest
    ss1 += __shfl_xor(ss1, 16);
    ss2 += __shfl_xor(ss2, 16);

    float d1 = c1[0];  // D[row][laneN] — rows identical, any row works
    float d2 = c2[0];

    float s1 = d1 * invq1 / fmaxf(sqrtf(ss1), 1e-12f);
    float s2 = d2 * invq2 / fmaxf(sqrtf(ss2), 1e-12f);
    s1 = fminf(fmaxf(s1, 0.0f), 1.0f);     // clip before softmax
    s2 = fminf(fmaxf(s2, 0.0f), 1.0f);

    float ev1 = expf(s1);                  // s in [0,1]: safe w/o max-sub
    float ev2 = expf(s2);

    bool valid = (hi == 0) && (key0 < N);  // half-wave holds unique keys
    if (valid) {
      e1[key0] = ev1;
      e2[key0] = ev2;
    }
    z1 = valid ? ev1 : 0.0f;
    z2 = valid ? ev2 : 0.0f;
  }

  // wave reduction of softmax-denominator contributions
  for (int m = 16; m >= 1; m >>= 1) {
    z1 += __shfl_xor(z1, m);
    z2 += __shfl_xor(z2, m);
  }
  if (lane == 0) { wred1[wave] = z1; wred2[wave] = z2; }
  __syncthreads();
  if (t == 0) {
    float a = 0.0f, b = 0.0f;
    for (int w = 0; w < 8; ++w) { a += wred1[w]; b += wred2[w]; }
    partials[2 * (size_t)blockIdx.x]     = a;
    partials[2 * (size_t)blockIdx.x + 1] = b;
  }
}

// ---------------------------------------------------------------------------
// Kernel 2: deterministic reduction of per-block partials -> Z1, Z2
// ---------------------------------------------------------------------------
__global__ void nb_reduce_z_kernel(const float* __restrict__ partials,
                                   int nPart, float* __restrict__ hdr) {
  __shared__ float r1[256];
  __shared__ float r2[256];
  int t = threadIdx.x;
  float s1 = 0.0f, s2 = 0.0f;
  for (int i = t; i < nPart; i += 256) {
    s1 += partials[2 * (size_t)i];
    s2 += partials[2 * (size_t)i + 1];
  }
  r1[t] = s1; r2[t] = s2;
  __syncthreads();
  for (int off = 128; off > 0; off >>= 1) {
    if (t < off) { r1[t] += r1[t + off]; r2[t] += r2[t + off]; }
    __syncthreads();
  }
  if (t == 0) { hdr[0] = r1[0]; hdr[1] = r2[0]; }
}

// ---------------------------------------------------------------------------
// Kernel 3: blend softmaxes -> attn (overwrites e1), zero d_out
// ---------------------------------------------------------------------------
__global__ void nb_blend_kernel(const float* __restrict__ hdr,
                                float* __restrict__ e1,   // becomes attn
                                const float* __restrict__ e2,
                                float* __restrict__ out, int N) {
  int i = blockIdx.x * blockDim.x + threadIdx.x;
  if (i < N) {
    float a  = hdr[4];
    float iZ1 = 1.0f / hdr[0];
    float iZ2 = 1.0f / hdr[1];
    float attn = a * e1[i] * iZ1 + (1.0f - a) * e2[i] * iZ2;
    e1[i] = attn;
    out[i] = 0.0f;
  }
}

// ---------------------------------------------------------------------------
// Kernel 4: per-chunk top-25 via LDS iterative argmax (first-index tiebreak)
// ---------------------------------------------------------------------------
__global__ void nb_topk_stageA(const float* __restrict__ attn, int N,
                               float* __restrict__ cand_val,
                               int* __restrict__ cand_idx) {
  __shared__ float sv[CHUNK];
  __shared__ float rv[256];
  __shared__ int   ri[256];
  int t = threadIdx.x;
  int base = blockIdx.x * CHUNK;
  for (int s = t; s < CHUNK; s += 256) {
    int g = base + s;
    sv[s] = (g < N) ? attn[g] : -1.0f;   // attn > 0, pads never selected
  }
  __syncthreads();
  for (int it = 0; it < TOPK; ++it) {
    float bv = -2.0f; int bi = CHUNK;
    for (int s = t; s < CHUNK; s += 256) {
      float v = sv[s];
      if (v > bv || (v == bv && s < bi)) { bv = v; bi = s; }
    }
    rv[t] = bv; ri[t] = bi;
    __syncthreads();
    for (int off = 128; off > 0; off >>= 1) {
      if (t < off) {
        float ov = rv[t + off]; int oi = ri[t + off];
        if (ov > rv[t] || (ov == rv[t] && oi < ri[t])) { rv[t] = ov; ri[t] = oi; }
      }
      __syncthreads();
    }
    if (t == 0) {
      cand_val[blockIdx.x * TOPK + it] = rv[0];
      cand_idx[blockIdx.x * TOPK + it] = base + ri[0];
      sv[ri[0]] = -2.0f;                 // remove; below pad value
    }
    __syncthreads();
  }
}

// ---------------------------------------------------------------------------
// Kernel 5: merge candidates -> final top-25; rank 0 (argmax) suppressed
// ---------------------------------------------------------------------------
__global__ void nb_topk_stageB(const float* __restrict__ cand_val,
                               const int* __restrict__ cand_idx,
                               int nCand, float* __restrict__ out) {
  __shared__ float sv[CHUNK];
  __shared__ int   si[CHUNK];
  __shared__ float rv[256];
  __shared__ int   ri[256];
  int t = threadIdx.x;
  for (int s = t; s < CHUNK; s += 256) {
    sv[s] = (s < nCand) ? cand_val[s] : -1.0f;
    si[s] = (s < nCand) ? cand_idx[s] : 0x7fffffff;
  }
  __syncthreads();
  for (int it = 0; it < TOPK; ++it) {
    float bv = -2.0f; int bs = CHUNK - 1;
    for (int s = t; s < CHUNK; s += 256) {
      float v = sv[s];
      if (v > bv || (v == bv && si[s] < si[bs])) { bv = v; bs = s; }
    }
    rv[t] = bv; ri[t] = bs;
    __syncthreads();
    for (int off = 128; off > 0; off >>= 1) {
      if (t < off) {
        float ov = rv[t + off]; int os = ri[t + off];
        if (ov > rv[t] || (ov == rv[t] && si[os] < si[ri[t]])) {
          rv[t] = ov; ri[t] = os;
        }
      }
      __syncthreads();
    }
    if (t == 0) {
      if (it > 0) out[si[ri[0]]] = rv[0];  // rank 0 == argmax -> masked to 0
      sv[ri[0]] = -2.0f;
    }
    __syncthreads();
  }
}

// ---------------------------------------------------------------------------
extern "C" void kernel_launch(void* const* d_in, const int* in_sizes, int n_in,
                              void* d_out, int out_size, void* d_ws, size_t ws_size,
                              hipStream_t stream) {
  const float* q1   = (const float*)d_in[0];
  const float* k1   = (const float*)d_in[1];
  const float* q2   = (const float*)d_in[2];
  const float* k2   = (const float*)d_in[3];
  const float* temp = (const float*)d_in[4];
  float* out = (float*)d_out;

  int D = in_sizes[0];       // 128
  int N = in_sizes[1] / D;   // 500000
  (void)n_in; (void)out_size; (void)ws_size;

  // workspace layout (floats)
  float* hdr = (float*)d_ws;                      // 64
  int numTiles = (N + 15) / 16;
  int blocks1  = (numTiles + 7) / 8;              // 8 waves / block
  float* partials = hdr + 64;                     // 2 * blocks1
  size_t off = (size_t)64 + 2 * (size_t)blocks1;
  off = (off + 15) & ~(size_t)15;
  float* e1 = hdr + off;                          // N (later: attn)
  float* e2 = e1 + (size_t)N;                     // N
  float* cand_val = e2 + (size_t)N;               // CHUNK
  int*   cand_idx = (int*)(cand_val + CHUNK);     // CHUNK

  int blocksA = (N + CHUNK - 1) / CHUNK;
  int nCand = blocksA * TOPK;                     // 3075 for N=500000

  nb_init_kernel<<<1, DQ, 0, stream>>>(q1, q2, temp, hdr);
  nb_scores_kernel<<<blocks1, 256, 0, stream>>>(q1, k1, q2, k2, hdr,
                                                e1, e2, partials, N);
  nb_reduce_z_kernel<<<1, 256, 0, stream>>>(partials, blocks1, hdr);
  nb_blend_kernel<<<(N + 255) / 256, 256, 0, stream>>>(hdr, e1, e2, out, N);
  nb_topk_stageA<<<blocksA, 256, 0, stream>>>(e1, N, cand_val, cand_idx);
  nb_topk_stageB<<<1, 256, 0, stream>>>(cand_val, cand_idx, nCand, out);
}